// GGAT1Layer_42984032698818
// MI455X (gfx1250) — compile-verified
//
#include <hip/hip_runtime.h>

// Problem constants (match reference)
#define NN 50000
#define EE 600000
#define DD 128
#define KK 25000   // ceil(0.5 * N)

typedef float v2f __attribute__((ext_vector_type(2)));
typedef float v8f __attribute__((ext_vector_type(8)));

// ---------------------------------------------------------------------------
// 1) Zero agg + rank, init keep/new_id
// ---------------------------------------------------------------------------
__global__ void init_kernel(float* __restrict__ agg,
                            int* __restrict__ keep,
                            int* __restrict__ new_id,
                            int* __restrict__ rank) {
    int idx = blockIdx.x * blockDim.x + threadIdx.x;
    if (idx < NN * DD) agg[idx] = 0.0f;
    if (idx < NN) { keep[idx] = 0; new_id[idx] = -1; rank[idx] = 0; }
}

// ---------------------------------------------------------------------------
// 2) agg[dst] += x[src]  (32 lanes per edge, float4 per lane)
// ---------------------------------------------------------------------------
__global__ void scatter_kernel(const float* __restrict__ x,
                               const int* __restrict__ src,
                               const int* __restrict__ dst,
                               float* __restrict__ agg) {
    unsigned t = blockIdx.x * blockDim.x + threadIdx.x;
    unsigned e = t >> 5;      // edge id
    unsigned q = t & 31u;     // which float4 of the 128-float row
    if (e >= EE) return;
    int s = src[e];
    int d = dst[e];
    float4 xv = ((const float4*)(x + (size_t)s * DD))[q];
    float* ap = agg + (size_t)d * DD + q * 4;
    atomicAdd(ap + 0, xv.x);
    atomicAdd(ap + 1, xv.y);
    atomicAdd(ap + 2, xv.z);
    atomicAdd(ap + 3, xv.w);
}

// ---------------------------------------------------------------------------
// 3) score[n] = tanh(agg[n].W1_rel + x[n].W1_root + b1); sortable radix key
//    One wave32 per node.
// ---------------------------------------------------------------------------
__global__ void score_kernel(const float* __restrict__ x,
                             const float* __restrict__ agg,
                             const float* __restrict__ W1_rel,
                             const float* __restrict__ W1_root,
                             const float* __restrict__ b1,
                             float* __restrict__ score,
                             unsigned* __restrict__ key) {
    int wave = threadIdx.x >> 5;
    int lane = threadIdx.x & 31;
    int n = blockIdx.x * (blockDim.x >> 5) + wave;
    if (n >= NN) return;
    const float* ar = agg + (size_t)n * DD;
    const float* xr = x   + (size_t)n * DD;
    float s = 0.0f;
    for (int i = lane; i < DD; i += 32)
        s = fmaf(ar[i], W1_rel[i], fmaf(xr[i], W1_root[i], s));
    for (int o = 16; o > 0; o >>= 1)
        s += __shfl_xor(s, o, 32);
    if (lane == 0) {
        float sc = tanhf(s + b1[0]);
        score[n] = sc;
        unsigned u = __float_as_uint(sc);
        // monotonic (ascending) unsigned key
        key[n] = (u & 0x80000000u) ? ~u : (u | 0x80000000u);
    }
}

// ---------------------------------------------------------------------------
// 4a) Sliced exact ranking:
//     rank[i] = #{j: key[j]>key[i]} + #{j<i: key[j]==key[i]}
//     N-dim split into SLICES, partial counts merged via atomicAdd -> 16x
//     shorter serial loop and ~3100 blocks of parallelism.
// ---------------------------------------------------------------------------
#define RANK_TILE 1024
#define SLICES    16
#define SLICE_LEN ((NN + SLICES - 1) / SLICES)   // 3125

__global__ void rank_partial_kernel(const unsigned* __restrict__ key,
                                    int* __restrict__ rank) {
    __shared__ unsigned sk[RANK_TILE];
    int i = blockIdx.x * blockDim.x + threadIdx.x;
    int begin = blockIdx.y * SLICE_LEN;
    int end = begin + SLICE_LEN;
    if (end > NN) end = NN;
    unsigned ki = (i < NN) ? key[i] : 0u;
    int cnt = 0;
    for (int base = begin; base < end; base += RANK_TILE) {
        int lim = end - base;
        if (lim > RANK_TILE) lim = RANK_TILE;
        __syncthreads();
        for (int j = threadIdx.x; j < lim; j += blockDim.x)
            sk[j] = key[base + j];
        __syncthreads();
        if (i < NN) {
            for (int j = 0; j < lim; ++j) {
                unsigned kj = sk[j];
                cnt += (int)((kj > ki) | ((kj == ki) & ((base + j) < i)));
            }
        }
    }
    if (i < NN && cnt) atomicAdd(&rank[i], cnt);
}

// ---------------------------------------------------------------------------
// 4b) Finalize: rank < K  =>  node kept; scatter perm / ids / outputs.
//     Ranks form a strict total order, so every slot 0..K-1 is written once.
// ---------------------------------------------------------------------------
__global__ void rank_finalize_kernel(const int* __restrict__ rank,
                                     const float* __restrict__ score,
                                     int* __restrict__ perm,
                                     int* __restrict__ keep,
                                     int* __restrict__ new_id,
                                     float* __restrict__ out_perm,
                                     float* __restrict__ out_scoreperm,
                                     float* __restrict__ out_batch) {
    int i = blockIdx.x * blockDim.x + threadIdx.x;
    if (i >= NN) return;
    int cnt = rank[i];
    if (cnt < KK) {
        perm[cnt]   = i;
        keep[i]     = 1;
        new_id[i]   = cnt;
        out_perm[cnt]      = (float)i;
        out_scoreperm[cnt] = score[i];
        out_batch[cnt]     = 0.0f;
    }
}

// ---------------------------------------------------------------------------
// 5) WMMA GEMM (fp32 exact):
//    x5[k,:] = x[perm[k],:] + (agg[perm[k]]@W2_rel + x[perm[k]]@W2_root + b2)*score[perm[k]]
//    A tile (16 rows of x and agg, 16 KB) staged in LDS once per block and
//    shared by all 8 waves (8x L2-traffic reduction); skip-connection row and
//    score also served from LDS. One 16x16 C tile per wave.
//    V_WMMA_F32_16X16X4_F32 accumulating over K=256.
// ---------------------------------------------------------------------------
__global__ void gemm_kernel(const float* __restrict__ x,
                            const float* __restrict__ agg,
                            const float* __restrict__ W2_rel,
                            const float* __restrict__ W2_root,
                            const float* __restrict__ b2,
                            const int* __restrict__ perm,
                            const float* __restrict__ score,
                            float* __restrict__ x5) {
    __shared__ float sx[16 * DD];       // x rows (also used for skip connection)
    __shared__ float sg[16 * DD];       // agg rows
    __shared__ float sscore[16];

    const int rt = blockIdx.x;          // row tile over K kept nodes

    if (threadIdx.x < 16) {
        int krow = rt * 16 + threadIdx.x;
        int r = (krow < KK) ? perm[krow] : 0;
        // stash source row id in sscore slot temporarily? no: load directly
        sscore[threadIdx.x] = score[r];
        // record row for cooperative loads via shared int array packed in sx tail?
        // simpler: recompute r in the load loop below (perm is L2-hot).
    }
    __syncthreads();

    // cooperative float4 loads: 16 rows * 32 float4s = 512 per matrix
    for (int t = threadIdx.x; t < 16 * (DD / 4); t += blockDim.x) {
        int row = t >> 5;               // /(DD/4)
        int c4  = t & 31;               // %(DD/4)
        int krow = rt * 16 + row;
        int r = (krow < KK) ? perm[krow] : 0;
        ((float4*)sx)[row * (DD / 4) + c4] =
            ((const float4*)(x + (size_t)r * DD))[c4];
        ((float4*)sg)[row * (DD / 4) + c4] =
            ((const float4*)(agg + (size_t)r * DD))[c4];
    }
    __syncthreads();

    const int wave = threadIdx.x >> 5;   // 0..7 : column tile
    const int lane = threadIdx.x & 31;
    const int m    = lane & 15;          // row-in-tile (A) / col-in-tile (B,C)
    const int half = lane >> 4;          // 0: K={0,1}, 1: K={2,3}
    const int koff = half * 2;
    const int ncol = wave * 16 + m;      // B load / C store column

    const float* xr = sx + m * DD;
    const float* gr = sg + m * DD;

    v8f acc = {};
    // x @ W2_root
    for (int kk = 0; kk < DD; kk += 4) {
        int ka = kk + koff;
        v2f a, b;
        a.x = xr[ka];
        a.y = xr[ka + 1];
        b.x = W2_root[(size_t)ka * DD + ncol];
        b.y = W2_root[(size_t)(ka + 1) * DD + ncol];
        acc = __builtin_amdgcn_wmma_f32_16x16x4_f32(
            false, a, false, b, (short)0, acc, false, false);
    }
    // agg @ W2_rel
    for (int kk = 0; kk < DD; kk += 4) {
        int ka = kk + koff;
        v2f a, b;
        a.x = gr[ka];
        a.y = gr[ka + 1];
        b.x = W2_rel[(size_t)ka * DD + ncol];
        b.y = W2_rel[(size_t)(ka + 1) * DD + ncol];
        acc = __builtin_amdgcn_wmma_f32_16x16x4_f32(
            false, a, false, b, (short)0, acc, false, false);
    }

    // C/D layout: VGPR v -> row (v + 8*half) in tile; column = lane%16 (+tile)
    float bias = b2[ncol];
    #pragma unroll
    for (int v = 0; v < 8; ++v) {
        int rowit = v + 8 * half;
        int kr = rt * 16 + rowit;
        if (kr < KK) {
            float val = sx[rowit * DD + ncol] + (acc[v] + bias) * sscore[rowit];
            x5[(size_t)kr * DD + ncol] = val;
        }
    }
}

// ---------------------------------------------------------------------------
// 6) Edge filtering / remap; all outputs written as float
// ---------------------------------------------------------------------------
__global__ void edge_kernel(const int* __restrict__ src,
                            const int* __restrict__ dst,
                            const float* __restrict__ attr,
                            const int* __restrict__ keep,
                            const int* __restrict__ new_id,
                            float* __restrict__ out_ei,
                            float* __restrict__ out_attr,
                            float* __restrict__ out_mask) {
    int e = blockIdx.x * blockDim.x + threadIdx.x;
    if (e >= EE) return;
    int s = src[e];
    int d = dst[e];
    int m = keep[s] & keep[d];
    out_ei[e]       = m ? (float)new_id[s] : -1.0f;
    out_ei[EE + e]  = m ? (float)new_id[d] : -1.0f;
    out_attr[e]     = m ? attr[e] : 0.0f;
    out_mask[e]     = (float)m;
}

// ---------------------------------------------------------------------------
// Launcher
// ---------------------------------------------------------------------------
static inline size_t align256(size_t v) { return (v + 255) & ~(size_t)255; }

extern "C" void kernel_launch(void* const* d_in, const int* in_sizes, int n_in,
                              void* d_out, int out_size, void* d_ws, size_t ws_size,
                              hipStream_t stream) {
    const float* x       = (const float*)d_in[0];
    const int*   ei      = (const int*)  d_in[1];   // [2,E] flat: src then dst
    const float* eattr   = (const float*)d_in[2];
    const float* W1_rel  = (const float*)d_in[3];
    const float* b1      = (const float*)d_in[4];
    const float* W1_root = (const float*)d_in[5];
    const float* W2_rel  = (const float*)d_in[6];
    const float* b2      = (const float*)d_in[7];
    const float* W2_root = (const float*)d_in[8];

    const int* src = ei;
    const int* dst = ei + EE;

    // Output layout (flat float32, reference return order)
    float* out           = (float*)d_out;
    float* out_x5        = out;                                 // K*128
    float* out_ei        = out_x5 + (size_t)KK * DD;            // 2*E
    float* out_attr      = out_ei + 2 * (size_t)EE;             // E
    float* out_batch     = out_attr + EE;                       // K
    float* out_perm      = out_batch + KK;                      // K
    float* out_scoreperm = out_perm + KK;                       // K
    float* out_mask      = out_scoreperm + KK;                  // E

    // Workspace carve-out
    char* ws = (char*)d_ws;
    float*    agg    = (float*)ws;    ws += align256((size_t)NN * DD * sizeof(float));
    float*    score  = (float*)ws;    ws += align256((size_t)NN * sizeof(float));
    unsigned* key    = (unsigned*)ws; ws += align256((size_t)NN * sizeof(unsigned));
    int*      perm   = (int*)ws;      ws += align256((size_t)KK * sizeof(int));
    int*      keep   = (int*)ws;      ws += align256((size_t)NN * sizeof(int));
    int*      new_id = (int*)ws;      ws += align256((size_t)NN * sizeof(int));
    int*      rank   = (int*)ws;      ws += align256((size_t)NN * sizeof(int));

    // 1) init
    {
        int total = NN * DD;
        init_kernel<<<(total + 255) / 256, 256, 0, stream>>>(agg, keep, new_id, rank);
    }
    // 2) scatter-add
    {
        long long total = (long long)EE * 32;
        scatter_kernel<<<(int)((total + 255) / 256), 256, 0, stream>>>(x, src, dst, agg);
    }
    // 3) score + keys (4 waves/block, 1 node per wave)
    {
        int nodes_per_block = 128 / 32;
        score_kernel<<<(NN + nodes_per_block - 1) / nodes_per_block, 128, 0, stream>>>(
            x, agg, W1_rel, W1_root, b1, score, key);
    }
    // 4) exact top-k ranking (sliced partial counts, then finalize)
    {
        dim3 grid((NN + 255) / 256, SLICES);
        rank_partial_kernel<<<grid, 256, 0, stream>>>(key, rank);
        rank_finalize_kernel<<<(NN + 255) / 256, 256, 0, stream>>>(
            rank, score, perm, keep, new_id, out_perm, out_scoreperm, out_batch);
    }
    // 5) WMMA GEMM + skip connection -> x5
    {
        int row_tiles = (KK + 15) / 16;   // 1563
        gemm_kernel<<<row_tiles, 256, 0, stream>>>(
            x, agg, W2_rel, W2_root, b2, perm, score, out_x5);
    }
    // 6) edge filtering
    {
        edge_kernel<<<(EE + 255) / 256, 256, 0, stream>>>(
            src, dst, eattr, keep, new_id, out_ei, out_attr, out_mask);
    }
}